// ResNet_48077863911813
// MI455X (gfx1250) — compile-verified
//
#include <hip/hip_runtime.h>

typedef __attribute__((ext_vector_type(2))) float v2f;
typedef __attribute__((ext_vector_type(8))) float v8f;

#define LTOT    336                      // true sequence length
#define LSTRIDE 352                      // padded row (11 * 32): strips never cross batches
#define BATCH   4
#define NCOL    (BATCH * LSTRIDE)        // 1408 columns (batch folded into N)
#define BUF_F   (128 * NCOL)             // 180224 floats per activation buffer
#define ITER_F  ((size_t)4 * BUF_F)      // 4 rotating buffers per iteration
#define NT32    44                       // 1408/32 N-strips
#define TPB     11                       // strips per batch
#define PRED    96
#define BN_EPS  1e-5f

struct ConvArgs {
  const float* x;                        // raw input [B,336,7] (used when slotIn < 0)
  const float* wp;                       // tap-padded weights [Cout][Cin][k4]
  const float* g; const float* bt; const float* mn; const float* vr;
  float* iterBase;                       // ws + weight region
  int slotIn;                            // -1 => read raw x (with suffix offset i)
  int slotOut;
  int slotRes;                           // -1 => none (pre-BN'd shortcut buffer)
  int i0, relu;
};

// 128-thread block = 4 waves sharing one staged input window.
// Each wave: one 16(co) x 32(n) output strip = 2 accumulators, 2 WMMAs/K-step.
// V_WMMA_F32_16X16X4_F32, full fp32. A: lanes0-15 row=lane K={0,1}; lanes16-31
// row=lane-16 K={2,3}. B mirrors the K split. C: vgpr r = rows r / r+8.
// K/CIN are compile-time so all tap/channel indexing strength-reduces.
template <int K, int CIN>
__global__ __launch_bounds__(128) void conv_bn_wmma(ConvArgs p) {
  constexpr int S4  = (K == 1) ? 0 : ((K <= 4) ? 2 : 3);
  constexpr int K4  = 1 << S4;           // taps padded to 1, 4 or 8
  constexpr int WP  = 32 + K4 - 1;       // staged window width
  constexpr int PAD = (K - 1) / 2;       // torch 'same' left pad
  constexpr int KK4 = CIN * K4;

  const int tid   = threadIdx.x;
  const int lane  = tid & 31;
  const int wv    = tid >> 5;
  const int it    = blockIdx.y;
  const int i     = p.i0 + it;
  const int Li    = LTOT - i;            // suffix length
  const int tcoB  = blockIdx.x / NT32;
  const int tn    = blockIdx.x % NT32;
  const int b     = tn / TPB;
  const int l0    = (tn % TPB) * 32;
  const int n0    = b * LSTRIDE + l0;
  const int co0   = (tcoB * 4 + wv) * 16;
  const int colC  = lane & 15;
  const int mbase = (lane >> 4) * 8;

  float* OUT = p.iterBase + (size_t)it * ITER_F + (size_t)p.slotOut * BUF_F;

  __shared__ float sIn[CIN * WP];

  v8f acc0 = {0.f,0.f,0.f,0.f,0.f,0.f,0.f,0.f};
  v8f acc1 = {0.f,0.f,0.f,0.f,0.f,0.f,0.f,0.f};

  if (l0 < Li) {                         // uniform per block
    // Stage input window [CIN][WP]; zeros outside the suffix [0, Li).
    const float* INB = (p.slotIn >= 0)
        ? (p.iterBase + (size_t)it * ITER_F + (size_t)p.slotIn * BUF_F) : nullptr;
    const int l_in0 = l0 - PAD;
    for (int idx = tid; idx < CIN * WP; idx += 128) {
      int c  = idx / WP;
      int dl = idx - c * WP;
      int li = l_in0 + dl;
      float v = 0.f;
      if (li >= 0 && li < Li) {
        if (INB) v = INB[(size_t)c * NCOL + b * LSTRIDE + li];
        else     v = p.x[((size_t)b * LTOT + (li + i)) * 7 + c];  // x[b, li+i, c]
      }
      sIn[c * WP + dl] = v;
    }
    __syncthreads();

    const size_t aRow4 = (size_t)(co0 + colC) * KK4;
    const int    kofs  = (lane < 16) ? 0 : 2;

    #pragma unroll 4
    for (int kc0 = 0; kc0 < KK4; kc0 += 4) {
      const int   ka  = kc0 + kofs;
      const int   kb  = ka + 1;
      const float a0  = p.wp[aRow4 + ka];      // zero on padded taps
      const float a1  = p.wp[aRow4 + kb];
      const int   bsa = (ka >> S4) * WP + (ka & (K4 - 1)) + colC;
      const int   bsb = (kb >> S4) * WP + (kb & (K4 - 1)) + colC;
      v2f av  = {a0, a1};
      v2f bv0 = {sIn[bsa],      sIn[bsb]};     // cols n0+colC
      v2f bv1 = {sIn[bsa + 16], sIn[bsb + 16]};// cols n0+16+colC
      acc0 = __builtin_amdgcn_wmma_f32_16x16x4_f32(false, av, false, bv0,
                                                   (short)0, acc0, false, false);
      acc1 = __builtin_amdgcn_wmma_f32_16x16x4_f32(false, av, false, bv1,
                                                   (short)0, acc1, false, false);
    }
  }

  const float* RES = (p.slotRes >= 0)
      ? (p.iterBase + (size_t)it * ITER_F + (size_t)p.slotRes * BUF_F) : nullptr;
  const int la = l0 + colC, lb = la + 16;
  const int na = n0 + colC, nb = na + 16;
  #pragma unroll
  for (int r = 0; r < 8; ++r) {
    const int   co = co0 + mbase + r;
    const float sc = rsqrtf(p.vr[co] + BN_EPS) * p.g[co];
    const float mu = p.mn[co];
    const float be = p.bt[co];
    float va = (acc0[r] - mu) * sc + be;
    float vb = (acc1[r] - mu) * sc + be;
    if (RES) {
      va += RES[(size_t)co * NCOL + na];
      vb += RES[(size_t)co * NCOL + nb];
    }
    if (p.relu) { va = fmaxf(va, 0.f); vb = fmaxf(vb, 0.f); }
    OUT[(size_t)co * NCOL + na] = (la >= Li) ? 0.f : va;
    OUT[(size_t)co * NCOL + nb] = (lb >= Li) ? 0.f : vb;
  }
}

// Expand [Cout][Cin][k] -> [Cout][Cin][k4] with zeros on taps t >= k.
struct PadArgs { const float* w; float* wp; int k, k4, total; };
__global__ void pad_weights(PadArgs p) {
  for (int idx = blockIdx.x * blockDim.x + threadIdx.x;
       idx < p.total; idx += gridDim.x * blockDim.x) {
    const int t  = idx % p.k4;
    const int rc = idx / p.k4;                 // co*Cin + ci
    p.wp[idx] = (t < p.k) ? p.w[rc * p.k + t] : 0.f;
  }
}

// Block-3 identity shortcut: s = BN(x), zero-masked past the suffix end.
__global__ void bn_shortcut(ConvArgs p) {
  const int it = blockIdx.y;
  const int i  = p.i0 + it;
  const int Li = LTOT - i;
  const float* IN = p.iterBase + (size_t)it * ITER_F + (size_t)p.slotIn * BUF_F;
  float* OUT      = p.iterBase + (size_t)it * ITER_F + (size_t)p.slotOut * BUF_F;
  for (int idx = blockIdx.x * blockDim.x + threadIdx.x;
       idx < 128 * NCOL; idx += gridDim.x * blockDim.x) {
    const int c = idx / NCOL;
    const int l = idx % LSTRIDE;               // NCOL = 4*LSTRIDE
    float v = 0.f;
    if (l < Li)
      v = (IN[idx] - p.mn[c]) * rsqrtf(p.vr[c] + BN_EPS) * p.g[c] + p.bt[c];
    OUT[idx] = v;
  }
}

struct HeadArgs {
  const float *w1, *b1, *w2, *b2;
  float* iterBase;
  float* out;
  int slotIn, i0;
};

// GAP over the suffix, then 128->7 and 7->7 FCs. One block per (iter, batch).
__global__ __launch_bounds__(128) void head_gap_fc(HeadArgs p) {
  const int b  = blockIdx.x;
  const int it = blockIdx.y;
  const int i  = p.i0 + it;
  const int Li = LTOT - i;
  const float* IN = p.iterBase + (size_t)it * ITER_F + (size_t)p.slotIn * BUF_F;
  __shared__ float sg[128];
  __shared__ float sh[7];
  const int c = threadIdx.x;
  const float* row = IN + (size_t)c * NCOL + b * LSTRIDE;
  float s = 0.f;
  for (int l = 0; l < Li; ++l) s += row[l];
  sg[c] = s / (float)Li;
  __syncthreads();
  if (c < 7) {
    float h = p.b1[c];
    for (int cc = 0; cc < 128; ++cc) h += sg[cc] * p.w1[cc * 7 + c];
    sh[c] = h;
  }
  __syncthreads();
  if (c < 7) {
    float o = p.b2[c];
    for (int j = 0; j < 7; ++j) o += sh[j] * p.w2[j * 7 + c];
    p.out[(size_t)b * PRED * 7 + (size_t)i * 7 + c] = o;
  }
}

extern "C" void kernel_launch(void* const* d_in, const int* in_sizes, int n_in,
                              void* d_out, int out_size, void* d_ws, size_t ws_size,
                              hipStream_t stream) {
  if (n_in < 64 || !d_out || !d_ws) return;

  // Harness flattens setup_inputs() in dict-insertion order (x first) or
  // jax-sorted order (x last, bn fields beta<gamma<mean<var). Detect by size.
  const bool ins = (in_sizes[0] == 9408);
  int X, W11, W12, W13, WS1, W21, W22, W23, WS2, W31, W32, W33;
  int B11, B12, B13, B1S, B21, B22, B23, B2S, B31, B32, B33, B3S;
  int FW1, FB1, FW2, FB2, GO, BO;
  if (ins) {
    X = 0;
    W11 = 1;  B11 = 2;  W12 = 6;  B12 = 7;  W13 = 11; B13 = 12; B1S = 16; WS1 = 20;
    W21 = 21; B21 = 22; W22 = 26; B22 = 27; W23 = 31; B23 = 32; B2S = 36; WS2 = 40;
    W31 = 41; B31 = 42; W32 = 46; B32 = 47; W33 = 51; B33 = 52; B3S = 56;
    FW1 = 60; FB1 = 61; FW2 = 62; FB2 = 63;
    GO = 0; BO = 1;                            // {gamma,beta,mean,var}
  } else {
    B11 = 0;  B12 = 4;  B13 = 8;  B1S = 12; W11 = 16; W12 = 17; W13 = 18; WS1 = 19;
    B21 = 20; B22 = 24; B23 = 28; B2S = 32; W21 = 36; W22 = 37; W23 = 38; WS2 = 39;
    B31 = 40; B32 = 44; B33 = 48; B3S = 52; W31 = 56; W32 = 57; W33 = 58;
    FB1 = 59; FB2 = 60; FW1 = 61; FW2 = 62; X = 63;
    GO = 1; BO = 0;                            // {beta,gamma,mean,var}
  }

  const float* xp = (const float*)d_in[X];
  float* wsf      = (float*)d_ws;
  float* outp     = (float*)d_out;
  auto F = [&](int idx) { return (const float*)d_in[idx]; };

  // ---- pad all conv weights into the front of the workspace ----
  // order: b1c1,b1c2,b1s,b1c3, b2c1,b2c2,b2s,b2c3, b3c1,b3c2,b3c3
  const int lw[11]  = {W11, W12, WS1, W13, W21, W22, WS2, W23, W31, W32, W33};
  const int lco[11] = {64, 64, 64, 64, 128, 128, 128, 128, 128, 128, 128};
  const int lci[11] = {7, 64, 7, 64, 64, 128, 64, 128, 128, 128, 128};
  const int lk[11]  = {8, 5, 1, 3, 8, 5, 1, 3, 8, 5, 3};
  auto k4of = [](int k) { return (k == 1) ? 1 : ((k <= 4) ? 4 : 8); };
  size_t woff[11];
  size_t wtot = 0;
  for (int j = 0; j < 11; ++j) {
    woff[j] = wtot;
    wtot += (size_t)lco[j] * lci[j] * k4of(lk[j]);
  }
  for (int j = 0; j < 11; ++j) {
    PadArgs pa;
    pa.w = F(lw[j]); pa.wp = wsf + woff[j];
    pa.k = lk[j]; pa.k4 = k4of(lk[j]);
    pa.total = lco[j] * lci[j] * pa.k4;
    pad_weights<<<dim3((pa.total + 255) / 256), dim3(256), 0, stream>>>(pa);
  }

  float* iterBase = wsf + wtot;                // activation region
  const size_t per_iter = ITER_F * sizeof(float);
  const size_t wbytes   = wtot * sizeof(float);
  int chunk = (ws_size > wbytes) ? (int)((ws_size - wbytes) / per_iter) : 1;
  if (chunk < 1) chunk = 1;
  if (chunk > PRED) chunk = PRED;

  for (int i0 = 0; i0 < PRED; i0 += chunk) {
    const int ni = (PRED - i0 < chunk) ? (PRED - i0) : chunk;

    auto conv = [&](int lyr, int bnI, int sIn, int sOut, int sRes,
                    int Cin, int Cout, int k, int relu) {
      ConvArgs a;
      a.x = xp; a.wp = wsf + woff[lyr];
      a.g = F(bnI + GO); a.bt = F(bnI + BO); a.mn = F(bnI + 2); a.vr = F(bnI + 3);
      a.iterBase = iterBase; a.slotIn = sIn; a.slotOut = sOut; a.slotRes = sRes;
      a.i0 = i0; a.relu = relu;
      dim3 grid((Cout / 64) * NT32, ni);       // 4 waves cover 64 co per block
      dim3 blk(128);
      if      (k == 8 && Cin == 7)   conv_bn_wmma<8, 7>  <<<grid, blk, 0, stream>>>(a);
      else if (k == 1 && Cin == 7)   conv_bn_wmma<1, 7>  <<<grid, blk, 0, stream>>>(a);
      else if (k == 5 && Cin == 64)  conv_bn_wmma<5, 64> <<<grid, blk, 0, stream>>>(a);
      else if (k == 3 && Cin == 64)  conv_bn_wmma<3, 64> <<<grid, blk, 0, stream>>>(a);
      else if (k == 8 && Cin == 64)  conv_bn_wmma<8, 64> <<<grid, blk, 0, stream>>>(a);
      else if (k == 1 && Cin == 64)  conv_bn_wmma<1, 64> <<<grid, blk, 0, stream>>>(a);
      else if (k == 5 && Cin == 128) conv_bn_wmma<5, 128><<<grid, blk, 0, stream>>>(a);
      else if (k == 3 && Cin == 128) conv_bn_wmma<3, 128><<<grid, blk, 0, stream>>>(a);
      else if (k == 8 && Cin == 128) conv_bn_wmma<8, 128><<<grid, blk, 0, stream>>>(a);
    };

    // block1 (7 -> 64)
    conv(0, B11, -1, 0, -1,  7,  64, 8, 1);
    conv(1, B12,  0, 1, -1, 64,  64, 5, 1);
    conv(2, B1S, -1, 2, -1,  7,  64, 1, 0);     // shortcut conv1x1 + BN
    conv(3, B13,  1, 3,  2, 64,  64, 3, 1);     // + residual + relu
    // block2 (64 -> 128)
    conv(4, B21,  3, 0, -1,  64, 128, 8, 1);
    conv(5, B22,  0, 1, -1, 128, 128, 5, 1);
    conv(6, B2S,  3, 2, -1,  64, 128, 1, 0);
    conv(7, B23,  1, 3,  2, 128, 128, 3, 1);
    // block3 (128 -> 128, identity shortcut)
    conv(8, B31,  3, 0, -1, 128, 128, 8, 1);
    conv(9, B32,  0, 1, -1, 128, 128, 5, 1);
    {
      ConvArgs a;
      a.x = xp; a.wp = nullptr;
      a.g = F(B3S + GO); a.bt = F(B3S + BO); a.mn = F(B3S + 2); a.vr = F(B3S + 3);
      a.iterBase = iterBase; a.slotIn = 3; a.slotOut = 2; a.slotRes = -1;
      a.i0 = i0; a.relu = 0;
      bn_shortcut<<<dim3(176, ni), dim3(256), 0, stream>>>(a);
    }
    conv(10, B33, 1, 0,  2, 128, 128, 3, 1);
    // GAP + FC head
    {
      HeadArgs h;
      h.w1 = F(FW1); h.b1 = F(FB1); h.w2 = F(FW2); h.b2 = F(FB2);
      h.iterBase = iterBase; h.out = outp; h.slotIn = 0; h.i0 = i0;
      head_gap_fc<<<dim3(BATCH, ni), dim3(128), 0, stream>>>(h);
    }
  }
  (void)out_size;
}